// Encoder_6914897347142
// MI455X (gfx1250) — compile-verified
//
#include <hip/hip_runtime.h>
#include <hip/hip_bf16.h>

typedef float v2f __attribute__((ext_vector_type(2)));
typedef float v8f __attribute__((ext_vector_type(8)));

#define NEG_SLOPE 0.2f

// ---------------- helpers ----------------
__device__ __forceinline__ float lrelu(float v) { return v > 0.f ? v : NEG_SLOPE * v; }

// monotone float -> uint map (for atomicMax-based segment max). map(real) > 0 always,
// so a 0-initialized buffer acts as -inf identity.
__device__ __forceinline__ unsigned fmap(float f) {
    unsigned u = __float_as_uint(f);
    return (u & 0x80000000u) ? ~u : (u | 0x80000000u);
}
__device__ __forceinline__ float funmap(unsigned u) {
    unsigned b = (u & 0x80000000u) ? (u & 0x7FFFFFFFu) : ~u;
    return __uint_as_float(b);
}

// ---------------- WMMA fp32 GEMM: C[N x M] = A[N x K] * B[K x M] ----------------
// One wave (block of 32) computes a 16(rows) x 64(cols) tile using
// v_wmma_f32_16x16x4_f32, reusing the A fragment across 4 B fragments.
// A-frag layout (16x4): lane<16 holds row=lane, K={0,1}; lane>=16 row=lane-16, K={2,3}.
// B-frag layout (4x16): VGPR v: lanes0-15 -> K=v, col=lane; lanes16-31 -> K=v+2.
// C/D layout (16x16):   VGPR i: lanes0-15 -> row=i; lanes16-31 -> row=i+8; col=lane&15.
__global__ void gat_gemm_wmma(const float* __restrict__ A, const float* __restrict__ B,
                              float* __restrict__ C, int N, int K, int M) {
    const int lane = threadIdx.x & 31;
    const int r    = lane & 15;
    const int hi   = lane >> 4;      // 0 or 1
    const int kh   = hi * 2;         // K sub-offset for this half-wave
    const int row0 = blockIdx.x * 16;
    const int col0 = blockIdx.y * 64;

    int arow = row0 + r;
    if (arow >= N) arow = N - 1;     // clamp (stores are guarded)
    const float* __restrict__ Arow = A + (size_t)arow * K;

    v8f acc0 = {}, acc1 = {}, acc2 = {}, acc3 = {};

    for (int k = 0; k < K; k += 4) {
        v2f a;
        a.x = Arow[k + kh];
        a.y = Arow[k + kh + 1];

        const float* __restrict__ B0 = B + (size_t)(k + kh) * M + col0 + r;
        const float* __restrict__ B1 = B0 + M;
        v2f b0, b1, b2, b3;
        b0.x = B0[0];  b0.y = B1[0];
        b1.x = B0[16]; b1.y = B1[16];
        b2.x = B0[32]; b2.y = B1[32];
        b3.x = B0[48]; b3.y = B1[48];

        acc0 = __builtin_amdgcn_wmma_f32_16x16x4_f32(false, a, false, b0, (short)0, acc0, false, false);
        acc1 = __builtin_amdgcn_wmma_f32_16x16x4_f32(false, a, false, b1, (short)0, acc1, false, false);
        acc2 = __builtin_amdgcn_wmma_f32_16x16x4_f32(false, a, false, b2, (short)0, acc2, false, false);
        acc3 = __builtin_amdgcn_wmma_f32_16x16x4_f32(false, a, false, b3, (short)0, acc3, false, false);
    }

    const int crow0 = row0 + hi * 8;
    const int ccol  = col0 + r;
    for (int i = 0; i < 8; ++i) {
        const int rr = crow0 + i;
        if (rr < N) {
            float* Cr = C + (size_t)rr * M + ccol;
            Cr[0]  = acc0[i];
            Cr[16] = acc1[i];
            Cr[32] = acc2[i];
            Cr[48] = acc3[i];
        }
    }
}

// ---------------- attention logit dot products: al[n,h] = <h[n,h,:], a[h,:]> ----------------
__global__ void gat_attn_logits(const float* __restrict__ hfeat,
                                const float* __restrict__ a_src,
                                const float* __restrict__ a_dst,
                                float* __restrict__ alS, float* __restrict__ alD,
                                int N, int H, int D) {
    int t = blockIdx.x * blockDim.x + threadIdx.x;
    if (t >= N * H) return;
    int n = t / H, hh = t % H;
    const float* row = hfeat + (size_t)n * H * D + (size_t)hh * D;
    const float* as  = a_src + hh * D;
    const float* ad  = a_dst + hh * D;
    float s0 = 0.f, s1 = 0.f;
    for (int d = 0; d < D; ++d) {
        float v = row[d];
        s0 = fmaf(v, as[d], s0);
        s1 = fmaf(v, ad[d], s1);
    }
    alS[t] = s0;
    alD[t] = s1;
}

// ---------------- edge pass 1: e = lrelu(alS[src]+alD[dst]); segment max into m ----------------
__global__ void gat_edge_logits_max(const int* __restrict__ ei, int E, int N, int H,
                                    const float* __restrict__ alS, const float* __restrict__ alD,
                                    float* __restrict__ elog, unsigned* __restrict__ m) {
    int e = blockIdx.x * blockDim.x + threadIdx.x;
    const int Et = E + N;
    if (e >= Et) return;
    int s, d;
    if (e < E) { s = ei[e]; d = ei[E + e]; } else { s = d = e - E; }
    for (int hh = 0; hh < H; ++hh) {
        float v = lrelu(alS[s * H + hh] + alD[d * H + hh]);
        elog[(size_t)e * H + hh] = v;
        atomicMax(&m[d * H + hh], fmap(v));
    }
}

// ---------------- edge pass 2: ex = exp(e - m[dst]); segment sum into den; store ex ----------------
__global__ void gat_edge_expsum(const int* __restrict__ ei, int E, int N, int H,
                                const unsigned* __restrict__ m,
                                float* __restrict__ elog, float* __restrict__ den) {
    int e = blockIdx.x * blockDim.x + threadIdx.x;
    const int Et = E + N;
    if (e >= Et) return;
    int d;
    if (e < E) { d = ei[E + e]; } else { d = e - E; }
    for (int hh = 0; hh < H; ++hh) {
        float ex = __expf(elog[(size_t)e * H + hh] - funmap(m[d * H + hh]));
        elog[(size_t)e * H + hh] = ex;   // overwrite logit with numerator
        atomicAdd(&den[d * H + hh], ex);
    }
}

// ---------------- edge pass 3: agg[dst] += (ex/den) * h[src]  (one wave per edge) ----------------
__global__ void gat_edge_aggregate(const int* __restrict__ ei, int E, int N, int H, int D,
                                   const float* __restrict__ hfeat,
                                   const float* __restrict__ elog,
                                   const float* __restrict__ den,
                                   float* __restrict__ agg) {
    const int wave = blockIdx.x * (blockDim.x >> 5) + (threadIdx.x >> 5);
    const int lane = threadIdx.x & 31;
    const int Et = E + N;
    if (wave >= Et) return;
    int s, d;
    if (wave < E) { s = ei[wave]; d = ei[E + wave]; } else { s = d = wave - E; }
    const int HD = H * D;
    for (int f = lane; f < HD; f += 32) {
        int hh = f / D;
        float alpha = elog[(size_t)wave * H + hh] / fmaxf(den[d * H + hh], 1e-16f);
        atomicAdd(&agg[(size_t)d * HD + f], alpha * hfeat[(size_t)s * HD + f]);
    }
}

// ---------------- x2 = leaky_relu(agg + b) ----------------
__global__ void gat_bias_act(const float* __restrict__ agg, const float* __restrict__ b,
                             float* __restrict__ outp, int total, int HD) {
    int t = blockIdx.x * blockDim.x + threadIdx.x;
    if (t >= total) return;
    outp[t] = lrelu(agg[t] + b[t % HD]);
}

// ---------------- out = mean_h(agg2) + b2 ----------------
__global__ void gat_head_mean_bias(const float* __restrict__ agg, const float* __restrict__ b,
                                   float* __restrict__ outp, int N, int H, int D) {
    int t = blockIdx.x * blockDim.x + threadIdx.x;
    if (t >= N * D) return;
    int n = t / D, d = t % D;
    const float* base = agg + (size_t)n * H * D + d;
    float s = 0.f;
    for (int hh = 0; hh < H; ++hh) s += base[hh * D];
    outp[t] = s * (1.f / (float)H) + b[d];
}

extern "C" void kernel_launch(void* const* d_in, const int* in_sizes, int n_in,
                              void* d_out, int out_size, void* d_ws, size_t ws_size,
                              hipStream_t stream) {
    const float* x   = (const float*)d_in[0];
    const int*   ei  = (const int*)d_in[1];   // edge_index materializes as int32 (default JAX x64-off)
    const float* W1  = (const float*)d_in[2];
    const float* aS1 = (const float*)d_in[3];
    const float* aD1 = (const float*)d_in[4];
    const float* b1  = (const float*)d_in[5];
    const float* W2  = (const float*)d_in[6];
    const float* aS2 = (const float*)d_in[7];
    const float* aD2 = (const float*)d_in[8];
    const float* b2  = (const float*)d_in[9];
    float* out = (float*)d_out;

    const int CIN = 256, H = 4, D1 = 64, D2 = 32;
    const int HD1 = H * D1, HD2 = H * D2;
    const int N  = in_sizes[0] / CIN;
    const int E  = in_sizes[1] / 2;
    const int Et = E + N;

    // workspace layout (fp32 words)
    float* ws = (float*)d_ws;
    size_t o = 0;
    float*    h1   = ws + o; o += (size_t)N * HD1;  // layer-1 features; reused as layer-2 input
    float*    agg1 = ws + o; o += (size_t)N * HD1;
    float*    h2   = ws + o; o += (size_t)N * HD2;
    float*    agg2 = ws + o; o += (size_t)N * HD2;
    float*    alS  = ws + o; o += (size_t)N * H;
    float*    alD  = ws + o; o += (size_t)N * H;
    unsigned* m    = (unsigned*)(ws + o); o += (size_t)N * H;
    float*    den  = ws + o; o += (size_t)N * H;
    float*    elog = ws + o; o += (size_t)Et * H;

    const int TB = 256;

    // ================= Layer 1 =================
    {
        dim3 g((N + 15) / 16, HD1 / 64);
        gat_gemm_wmma<<<g, 32, 0, stream>>>(x, W1, h1, N, CIN, HD1);
    }
    gat_attn_logits<<<(N * H + TB - 1) / TB, TB, 0, stream>>>(h1, aS1, aD1, alS, alD, N, H, D1);
    hipMemsetAsync(m,    0, (size_t)N * H   * sizeof(float), stream);
    hipMemsetAsync(den,  0, (size_t)N * H   * sizeof(float), stream);
    hipMemsetAsync(agg1, 0, (size_t)N * HD1 * sizeof(float), stream);
    gat_edge_logits_max<<<(Et + TB - 1) / TB, TB, 0, stream>>>(ei, E, N, H, alS, alD, elog, m);
    gat_edge_expsum<<<(Et + TB - 1) / TB, TB, 0, stream>>>(ei, E, N, H, m, elog, den);
    gat_edge_aggregate<<<(Et + 7) / 8, TB, 0, stream>>>(ei, E, N, H, D1, h1, elog, den, agg1);
    gat_bias_act<<<((size_t)N * HD1 + TB - 1) / TB, TB, 0, stream>>>(agg1, b1, h1, N * HD1, HD1);

    // ================= Layer 2 =================
    {
        dim3 g((N + 15) / 16, HD2 / 64);
        gat_gemm_wmma<<<g, 32, 0, stream>>>(h1, W2, h2, N, HD1, HD2);
    }
    gat_attn_logits<<<(N * H + TB - 1) / TB, TB, 0, stream>>>(h2, aS2, aD2, alS, alD, N, H, D2);
    hipMemsetAsync(m,    0, (size_t)N * H   * sizeof(float), stream);
    hipMemsetAsync(den,  0, (size_t)N * H   * sizeof(float), stream);
    hipMemsetAsync(agg2, 0, (size_t)N * HD2 * sizeof(float), stream);
    gat_edge_logits_max<<<(Et + TB - 1) / TB, TB, 0, stream>>>(ei, E, N, H, alS, alD, elog, m);
    gat_edge_expsum<<<(Et + TB - 1) / TB, TB, 0, stream>>>(ei, E, N, H, m, elog, den);
    gat_edge_aggregate<<<(Et + 7) / 8, TB, 0, stream>>>(ei, E, N, H, D2, h2, elog, den, agg2);
    gat_head_mean_bias<<<(N * D2 + TB - 1) / TB, TB, 0, stream>>>(agg2, b2, out, N, H, D2);
}